// MCAO_29978871726232
// MI455X (gfx1250) — compile-verified
//
#include <hip/hip_runtime.h>

// Problem constants (from the reference)
#define B_      4
#define T_      8192
#define F_      64
#define KTAPS   51        // MAX_K + 1
#define TILE_T  32
#define HALO    50        // KTAPS - 1
#define ROWS    (TILE_T + HALO)   // 82
#define PAD     65        // LDS row pitch (kills stride-64 bank conflicts)

typedef __attribute__((ext_vector_type(2))) float v2f;
typedef __attribute__((ext_vector_type(8))) float v8f;

__device__ __forceinline__ float fast_tanh(float v) {
#if __has_builtin(__builtin_amdgcn_tanhf)
    return __builtin_amdgcn_tanhf(v);          // v_tanh_f32 (gfx1250 trans op)
#elif __has_builtin(__builtin_amdgcn_tanh_f32)
    return __builtin_amdgcn_tanh_f32(v);
#else
    return tanhf(v);
#endif
}

__device__ __forceinline__ float fast_rcp(float v) {
#if __has_builtin(__builtin_amdgcn_rcpf)
    return __builtin_amdgcn_rcpf(v);           // v_rcp_f32
#else
    return 1.0f / v;
#endif
}

// --- Kernel 1: row softmax of the 64x64 weight matrix into workspace ---
__global__ __launch_bounds__(64) void MCAO_softmax64(const float* __restrict__ wm,
                                                     float* __restrict__ wout) {
    const int r = threadIdx.x;           // one thread per row, 64 threads
    float m = -3.4e38f;
    for (int j = 0; j < 64; ++j) m = fmaxf(m, wm[r * 64 + j]);
    float s = 0.0f;
    for (int j = 0; j < 64; ++j) s += __expf(wm[r * 64 + j] - m);
    const float inv = 1.0f / s;
    for (int j = 0; j < 64; ++j) wout[r * 64 + j] = __expf(wm[r * 64 + j] - m) * inv;
}

// --- Kernel 2: fused memory/coupling/global-term kernel ---
// grid = B * (T/TILE_T) blocks, 256 threads (8 wave32) each.
// Wave w owns the 16x16 proj tile (tm = w>>2, tn = w&3) over (t-row, feature).
__global__ __launch_bounds__(256) void MCAO_fused(
    const float* __restrict__ x,      // [B,T,F]
    const float* __restrict__ wsm,    // [F,F] softmaxed weights
    const float* __restrict__ pw,     // [F,F] proj_w
    const float* __restrict__ pb,     // [F]
    const float* __restrict__ kappa,  // [1]
    const float* __restrict__ coeffs, // [KTAPS]
    float* __restrict__ outp,         // [B,T,F] output
    float* __restrict__ memp)         // [B,T,F] memory_term
{
    __shared__ float  sX[ROWS * PAD];     // x tile with causal halo      (~21.3 KB)
    __shared__ float  sW[64 * PAD];       // softmax(W), padded rows      (~16.6 KB)
    __shared__ float2 sTE[TILE_T * PAD];  // (tanh(x), exp(-|x|)) table   (~16.6 KB)
    __shared__ float  sC[64];             // conv coeffs

    const int tid  = threadIdx.x;
    const int bIdx = blockIdx.x / (T_ / TILE_T);
    const int t0   = (blockIdx.x % (T_ / TILE_T)) * TILE_T;

    const int wave  = tid >> 5;
    const int lane  = tid & 31;
    const int m16   = lane & 15;
    const int khalf = (lane < 16) ? 0 : 2;   // f32 A/B operand: K split by half-wave
    const int tm    = wave >> 2;             // t-tile   (rows r = tm*16 .. +15)
    const int tn    = wave & 3;              // i-tile   (cols i = tn*16 .. +15)

    // --- Preload B fragments (proj_w rows) into registers; the global loads
    //     overlap the LDS staging below so the WMMA chains run load-free.
    v2f bfr[16];
    {
        const float* prow = pw + (size_t)(tn * 16 + m16) * F_;
        #pragma unroll
        for (int kk = 0; kk < 16; ++kk) {
            const int kb = kk * 4 + khalf;
            bfr[kk].x = prow[kb];
            bfr[kk].y = prow[kb + 1];
        }
    }

    // Cooperative load of x tile rows [t0-50, t0+31], zero-pad t<0 (causal).
    // For the 32 output rows, also build the (tanh, exp(-|x|)) factor table
    // while the value is live in a register.
    for (int e = tid; e < ROWS * 64; e += 256) {
        const int j = e >> 6, f = e & 63;
        const int t = t0 - HALO + j;
        const float val = (t >= 0) ? x[((size_t)bIdx * T_ + t) * F_ + f] : 0.0f;
        sX[j * PAD + f] = val;
        if (j >= HALO) {
            float2 te;
            te.x = fast_tanh(val);
            te.y = __expf(-fabsf(val));
            sTE[(j - HALO) * PAD + f] = te;
        }
    }
    for (int e = tid; e < 64 * 64; e += 256)
        sW[(e >> 6) * PAD + (e & 63)] = wsm[e];
    if (tid < KTAPS) sC[tid] = coeffs[tid];
    __syncthreads();

    const float kap    = kappa[0];
    const float inv1mr = 1.0f / (1.0f - __expf(-kap));   // 1/(1-r), r=exp(-kappa)

    // --- A fragments from LDS ---
    v2f afr[16];
    {
        const int arow = HALO + tm * 16 + m16;
        #pragma unroll
        for (int kk = 0; kk < 16; ++kk) {
            const int kb = kk * 4 + khalf;
            afr[kk].x = sX[arow * PAD + kb];
            afr[kk].y = sX[arow * PAD + kb + 1];
        }
    }

    // --- Two independent WMMA chains (even/odd K split), summed at the end ---
    v8f accE = {0.f, 0.f, 0.f, 0.f, 0.f, 0.f, 0.f, 0.f};
    v8f accO = {0.f, 0.f, 0.f, 0.f, 0.f, 0.f, 0.f, 0.f};
    #pragma unroll
    for (int kk = 0; kk < 16; kk += 2) {
        accE = __builtin_amdgcn_wmma_f32_16x16x4_f32(
            false, afr[kk],     false, bfr[kk],     (short)0, accE, false, false);
        accO = __builtin_amdgcn_wmma_f32_16x16x4_f32(
            false, afr[kk + 1], false, bfr[kk + 1], (short)0, accO, false, false);
    }
    v8f acc = accE + accO;

    // --- Per-element epilogue ---
    const int   i    = tn * 16 + m16;            // N = lane%16 (D-matrix layout)
    const float bias = pb[i];
    #pragma unroll 2
    for (int v = 0; v < 8; ++v) {
        const int r = tm * 16 + v + ((lane >= 16) ? 8 : 0);  // M = v + 8*(lane>=16)
        const int t = t0 + r;

        // this element's own factors
        const float2 tei = sTE[r * PAD + i];
        const float  ta  = tei.x;                // tanh(xi)
        const float  ei  = tei.y;                // exp(-|xi|)

        // 51-tap causal conv (halo rows are zeroed for t<0)
        float mem = 0.0f;
        #pragma unroll
        for (int k = 0; k < KTAPS; ++k)
            mem = fmaf(sC[k], sX[(HALO + r - k) * PAD + i], mem);

        // coupling via factored identities:
        //   tanh(xi-xj)            = (ta - tb) / (1 - ta*tb)
        //   sigmoid(|xi| + |xj|)   = 1 / (1 + ei*ej)
        float coup = 0.0f;
        for (int j = 0; j < 64; ++j) {
            const float2 tej = sTE[r * PAD + j];           // broadcast b64 load
            const float  tb  = tej.x;
            const float  ej  = tej.y;
            const float  den = fmaxf(1.0f - ta * tb, 1e-20f);
            const float  th  = (ta - tb) * fast_rcp(den);
            const float  sg  = fast_rcp(1.0f + ei * ej);
            coup = fmaf(th * sg, sW[i * PAD + j], coup);
        }

        // cumsum(exp(-kappa*n), n<=t) closed form: (1 - r^(t+1)) / (1 - r)
        const float S    = (1.0f - __expf(-kap * (float)(t + 1))) * inv1mr;
        const float proj = acc[v] + bias;
        const float o    = 0.3f * mem + 0.4f * coup + 0.3f * proj * S;

        const size_t idx = ((size_t)bIdx * T_ + t) * F_ + i;
        outp[idx] = o;
        memp[idx] = mem;
    }
}

extern "C" void kernel_launch(void* const* d_in, const int* in_sizes, int n_in,
                              void* d_out, int out_size, void* d_ws, size_t ws_size,
                              hipStream_t stream) {
    const float* x   = (const float*)d_in[0];
    const float* wm  = (const float*)d_in[1];
    const float* pw  = (const float*)d_in[2];
    const float* pb  = (const float*)d_in[3];
    const float* kap = (const float*)d_in[4];
    const float* cf  = (const float*)d_in[5];

    float* wsm  = (float*)d_ws;                      // 64*64 softmaxed W
    float* outp = (float*)d_out;                     // output  [B,T,F]
    float* memp = outp + (size_t)B_ * T_ * F_;       // memory_term [B,T,F]

    MCAO_softmax64<<<1, 64, 0, stream>>>(wm, wsm);
    MCAO_fused<<<B_ * (T_ / TILE_T), 256, 0, stream>>>(x, wsm, pw, pb, kap, cf,
                                                       outp, memp);
}